// Unpooling2D_35570919145830
// MI455X (gfx1250) — compile-verified
//
#include <hip/hip_runtime.h>

// Max-unpooling with switches, POOL=(2,2), STRIDES=(2,2) -> non-overlapping
// windows, so the reference's scatter-add + mask-divide reduces exactly to:
//   out[b,h,w,c] = (pool_input[b,h,w,c] == max of its 2x2 window)
//                    ? input_tensor[b,h/2,w/2,c] : 0
// Shapes: input_tensor (32,64,64,64) f32, pool_input/out (32,128,128,64) f32.
// Pure streaming op: ~302 MB of traffic, zero reuse -> HBM-bound (~13 us at
// 23.3 TB/s). Strategy: one thread per (b,i,j,c4) window-column group, all
// traffic as 128-bit non-temporal global ops (gfx1250 TH_NT cache policy).

typedef float v4f __attribute__((ext_vector_type(4)));

#define B    32
#define H    128   // pool_input height
#define Wd   128   // pool_input width
#define CH   64
#define HO   64
#define WO   64
#define C4   (CH / 4)          // 16 float4 groups per channel row
#define ROWF (Wd * CH)         // floats per pool_input row = 8192

__global__ __launch_bounds__(256)
void unpool_switch_kernel(const float* __restrict__ in_t,     // (32,64,64,64)
                          const float* __restrict__ pool_in,  // (32,128,128,64)
                          float* __restrict__ out) {          // (32,128,128,64)
    unsigned t  = blockIdx.x * 256u + threadIdx.x;
    unsigned c4 = t & 15u;          // channel group (4 floats)  bits [3:0]
    unsigned j  = (t >> 4) & 63u;   // pooled col               bits [9:4]
    unsigned i  = (t >> 10) & 63u;  // pooled row               bits [15:10]
    unsigned b  = t >> 16;          // batch                    bits [20:16]

    // top-left element of the 2x2 window (float offset)
    size_t base = (((size_t)b * H + 2u * i) * Wd + 2u * j) * CH + c4 * 4u;

    v4f v00 = __builtin_nontemporal_load((const v4f*)(pool_in + base));
    v4f v01 = __builtin_nontemporal_load((const v4f*)(pool_in + base + CH));
    v4f v10 = __builtin_nontemporal_load((const v4f*)(pool_in + base + ROWF));
    v4f v11 = __builtin_nontemporal_load((const v4f*)(pool_in + base + ROWF + CH));

    size_t gidx = (((size_t)b * HO + i) * WO + j) * CH + c4 * 4u;
    v4f g = __builtin_nontemporal_load((const v4f*)(in_t + gidx));

    v4f r00, r01, r10, r11;
#pragma unroll
    for (int k = 0; k < 4; ++k) {
        float m = fmaxf(fmaxf(v00[k], v01[k]), fmaxf(v10[k], v11[k]));
        r00[k] = (v00[k] == m) ? g[k] : 0.0f;
        r01[k] = (v01[k] == m) ? g[k] : 0.0f;
        r10[k] = (v10[k] == m) ? g[k] : 0.0f;
        r11[k] = (v11[k] == m) ? g[k] : 0.0f;
    }

    __builtin_nontemporal_store(r00, (v4f*)(out + base));
    __builtin_nontemporal_store(r01, (v4f*)(out + base + CH));
    __builtin_nontemporal_store(r10, (v4f*)(out + base + ROWF));
    __builtin_nontemporal_store(r11, (v4f*)(out + base + ROWF + CH));
}

extern "C" void kernel_launch(void* const* d_in, const int* in_sizes, int n_in,
                              void* d_out, int out_size, void* d_ws, size_t ws_size,
                              hipStream_t stream) {
    (void)in_sizes; (void)n_in; (void)out_size; (void)d_ws; (void)ws_size;
    const float* in_t = (const float*)d_in[0];   // input_tensor (32,64,64,64)
    const float* pool = (const float*)d_in[1];   // pool_input   (32,128,128,64)
    float*       outp = (float*)d_out;           // (32,128,128,64)

    const unsigned total = (unsigned)B * HO * WO * C4;  // 2,097,152 threads
    dim3 grid(total / 256u), block(256u);
    hipLaunchKernelGGL(unpool_switch_kernel, grid, block, 0, stream,
                       in_t, pool, outp);
}